// RawWaveformMamba_79216376807622
// MI455X (gfx1250) — compile-verified
//
#include <hip/hip_runtime.h>
#include <hip/hip_bf16.h>
#include <math.h>

#define D_MODEL 256
#define D_STATE 32
#define D_CONV  4
#define DI      512
#define DT_RANK 16
#define N_LAYERS 6
#define PATCH   160
#define NCLS    50
#define BSZ     4
#define SEQ     512            // 81920 / 160
#define BL      (BSZ * SEQ)    // 2048 rows
#define MREP    4              // M-tiles per wave (64 rows)

typedef __attribute__((ext_vector_type(2))) float v2f;
typedef __attribute__((ext_vector_type(8))) float v8f;

__device__ __forceinline__ float siluf(float x) { return x / (1.f + __expf(-x)); }

// ---------------------------------------------------------------------------
// GEMM: C[M,N] = act( A[M,K] @ W[K,N] + bias + res )   (f32 WMMA 16x16x4)
// One wave32 computes a 64x16 tile of C (4 M-tiles sharing the B fragment),
// stepping K by 4. Per K-step: loads are issued as one batch (A x4 + B x2),
// then 4 WMMAs run back-to-back on independent accumulators.
// Fast path (full tiles): unconditional loads, no exec manipulation.
// Slow path (edge tiles): clamped addresses + multiply masks; loads stay
// unconditional and EXEC stays all-1s for the WMMAs (ISA requirement).
// A frag (16x4):  lane&15 = M row; vgpr0 = K = 2*(lane>>4); vgpr1 = K+1
// B frag (4x16):  lane&15 = N col; vgpr0 = K = 2*(lane>>4); vgpr1 = K+1
// C/D   (16x16):  vgpr r -> M = r + 8*(lane>=16), N = lane&15
// act: 0 = none, 1 = softplus
// ---------------------------------------------------------------------------
__global__ void __launch_bounds__(256)
gemm_f32_wmma(const float* __restrict__ A, const float* __restrict__ W,
              const float* __restrict__ bias, const float* __restrict__ res,
              float* __restrict__ C, int M, int N, int K,
              int lda, int ldb, int ldc, int act)
{
  const int lane   = threadIdx.x & 31;
  const int wave   = blockIdx.x * (blockDim.x >> 5) + (threadIdx.x >> 5);
  const int ntiles = (N + 15) >> 4;
  const int mt = wave / ntiles;
  const int nt = wave - mt * ntiles;
  const int m0 = mt * (MREP * 16), n0 = nt << 4;
  if (m0 >= M) return;                           // wave-uniform exit

  const int mr = lane & 15;
  const int kk = (lane >> 4) << 1;               // 0 or 2
  const int bn = n0 + (lane & 15);

  v8f acc[MREP] = {};
  const bool full = (m0 + MREP * 16 <= M) && (n0 + 16 <= N);  // wave-uniform

  if (full) {
    const float* Ap = A + (size_t)(m0 + mr) * lda + kk;
    const float* Wp = W + (size_t)kk * ldb + bn;
    const size_t arow16 = (size_t)16 * lda;
    for (int k0 = 0; k0 < K; k0 += 4) {
      // ---- load phase: one clause of independent loads ----
      v2f a[MREP];
      #pragma unroll
      for (int t = 0; t < MREP; ++t)
        a[t] = *(const v2f*)(Ap + (size_t)t * arow16);
      v2f b;
      b.x = Wp[0];
      b.y = Wp[(size_t)ldb];
      __builtin_prefetch(Wp + (size_t)4 * ldb, 0, 1);   // speculative: safe
      // ---- compute phase: 4 back-to-back WMMAs ----
      #pragma unroll
      for (int t = 0; t < MREP; ++t)
        acc[t] = __builtin_amdgcn_wmma_f32_16x16x4_f32(false, a[t], false, b,
                                                       (short)0, acc[t],
                                                       false, false);
      Ap += 4;
      Wp += (size_t)4 * ldb;
    }
    const float bv = bias ? bias[bn] : 0.f;
    #pragma unroll
    for (int t = 0; t < MREP; ++t) {
      const int mbase = m0 + t * 16 + ((lane >> 4) << 3);
      #pragma unroll
      for (int r = 0; r < 8; ++r) {
        const int m = mbase + r;
        float v = acc[t][r] + bv;
        if (res) v += res[(size_t)m * ldc + bn];
        if (act == 1) v = (v > 20.f) ? v : log1pf(__expf(v));  // softplus
        C[(size_t)m * ldc + bn] = v;
      }
    }
  } else {
    const int   bncl  = (bn < N) ? bn : (N - 1);
    const float bmask = (bn < N) ? 1.f : 0.f;
    size_t aoff[MREP];
    float  amask[MREP];
    #pragma unroll
    for (int t = 0; t < MREP; ++t) {
      const int row   = m0 + t * 16 + mr;
      amask[t]        = (row < M) ? 1.f : 0.f;
      const int rowcl = (row < M) ? row : (M - 1);
      aoff[t] = (size_t)rowcl * lda + kk;
    }
    const float* Wp = W + (size_t)kk * ldb + bncl;
    for (int k0 = 0; k0 < K; k0 += 4) {
      // ---- load phase ----
      v2f a[MREP];
      #pragma unroll
      for (int t = 0; t < MREP; ++t)
        a[t] = *(const v2f*)(A + aoff[t] + k0);
      v2f b;
      b.x = Wp[0];
      b.y = Wp[(size_t)ldb];
      // ---- mask + compute phase ----
      b.x *= bmask;
      b.y *= bmask;
      #pragma unroll
      for (int t = 0; t < MREP; ++t) {
        a[t].x *= amask[t];
        a[t].y *= amask[t];
      }
      #pragma unroll
      for (int t = 0; t < MREP; ++t)
        acc[t] = __builtin_amdgcn_wmma_f32_16x16x4_f32(false, a[t], false, b,
                                                       (short)0, acc[t],
                                                       false, false);
      Wp += (size_t)4 * ldb;
    }
    if (bn < N) {
      const float bv = bias ? bias[bn] : 0.f;
      #pragma unroll
      for (int t = 0; t < MREP; ++t) {
        const int mbase = m0 + t * 16 + ((lane >> 4) << 3);
        #pragma unroll
        for (int r = 0; r < 8; ++r) {
          const int m = mbase + r;
          if (m < M) {
            float v = acc[t][r] + bv;
            if (res) v += res[(size_t)m * ldc + bn];
            if (act == 1) v = (v > 20.f) ? v : log1pf(__expf(v));
            C[(size_t)m * ldc + bn] = v;
          }
        }
      }
    }
  }
}

// ---------------------------------------------------------------------------
// Patch embedding: out[b,l,d] = patch_b[d] + sum_k x[b, l*160+k] * pw[d,k]
// One block per (b,l); x segment staged in LDS, shared by all 256 d threads.
// ---------------------------------------------------------------------------
__global__ void patch_embed(const float* __restrict__ x, const float* __restrict__ pw,
                            const float* __restrict__ pb, float* __restrict__ out)
{
  __shared__ float sx[PATCH];
  const int bl = blockIdx.x;
  const int d  = threadIdx.x;
  const int b  = bl / SEQ, l = bl - b * SEQ;
  for (int k = d; k < PATCH; k += blockDim.x)
    sx[k] = x[(size_t)b * (SEQ * PATCH) + (size_t)l * PATCH + k];
  __syncthreads();
  float acc = pb[d];
  const float* w = pw + (size_t)d * PATCH;
  #pragma unroll 4
  for (int k = 0; k < PATCH; ++k) acc += sx[k] * w[k];
  out[(size_t)bl * D_MODEL + d] = acc;
}

// ---------------------------------------------------------------------------
// LayerNorm over last dim (256). One block (256 threads) per row.
// ---------------------------------------------------------------------------
__global__ void layernorm_k(const float* __restrict__ in, const float* __restrict__ g,
                            const float* __restrict__ b, float* __restrict__ out)
{
  __shared__ float red[D_MODEL];
  const int row = blockIdx.x, tid = threadIdx.x;
  const float v = in[(size_t)row * D_MODEL + tid];
  red[tid] = v; __syncthreads();
  for (int s = D_MODEL / 2; s > 0; s >>= 1) {
    if (tid < s) red[tid] += red[tid + s];
    __syncthreads();
  }
  const float mu = red[0] * (1.f / D_MODEL);
  __syncthreads();
  const float c = v - mu;
  red[tid] = c * c; __syncthreads();
  for (int s = D_MODEL / 2; s > 0; s >>= 1) {
    if (tid < s) red[tid] += red[tid + s];
    __syncthreads();
  }
  const float var = red[0] * (1.f / D_MODEL);
  out[(size_t)row * D_MODEL + tid] = c * rsqrtf(var + 1e-5f) * g[tid] + b[tid];
}

// ---------------------------------------------------------------------------
// Depthwise causal conv (K=4, left pad 3) + bias + SiLU, over xz[:, :DI].
// rev=1: time-reversed sequence (taps look forward) for the backward branch.
// ---------------------------------------------------------------------------
__global__ void conv_silu(const float* __restrict__ xz, const float* __restrict__ cw,
                          const float* __restrict__ cb, float* __restrict__ xc, int rev)
{
  const int idx = blockIdx.x * blockDim.x + threadIdx.x;   // BL*DI
  const int c  = idx & (DI - 1);
  const int bl = idx >> 9;
  const int b  = bl / SEQ, l = bl - b * SEQ;
  const float* w = cw + (size_t)c * D_CONV;
  float acc = cb[c];
  #pragma unroll
  for (int k = 0; k < D_CONV; ++k) {
    const int ls = rev ? (l + (D_CONV - 1) - k) : (l - (D_CONV - 1) + k);
    if (ls >= 0 && ls < SEQ)
      acc += w[k] * xz[(size_t)(b * SEQ + ls) * (2 * DI) + c];
  }
  xc[(size_t)bl * DI + c] = siluf(acc);
}

// ---------------------------------------------------------------------------
// Selective scan, fused with y = (ys + u*Dv) * silu(z).
// Single-wave blocks (32 threads): barriers become hardware S_NOPs.
// Grid (DI/32, BSZ) = 64 blocks; each lane owns one channel d, 32-wide state
// in registers; B_t / C_t staged in LDS per step (broadcast reads).
// rev=1 scans l = SEQ-1 .. 0 (backward branch).
// ---------------------------------------------------------------------------
__global__ void __launch_bounds__(32)
scan_k(const float* __restrict__ dt, const float* __restrict__ xc,
       const float* __restrict__ xdbl, const float* __restrict__ xz,
       const float* __restrict__ A_log, const float* __restrict__ Dv,
       float* __restrict__ y, int rev)
{
  const int d = blockIdx.x * 32 + threadIdx.x;   // 0..DI-1
  const int b = blockIdx.y;
  __shared__ float sB[D_STATE], sC[D_STATE];
  float Av[D_STATE], hc[D_STATE];
  #pragma unroll
  for (int n = 0; n < D_STATE; ++n) {
    Av[n] = -__expf(A_log[(size_t)d * D_STATE + n]);
    hc[n] = 0.f;
  }
  const float Dd = Dv[d];
  for (int s = 0; s < SEQ; ++s) {
    const int l   = rev ? (SEQ - 1 - s) : s;
    const int row = b * SEQ + l;
    __syncthreads();
    sB[threadIdx.x] = xdbl[(size_t)row * 80 + DT_RANK + threadIdx.x];
    sC[threadIdx.x] = xdbl[(size_t)row * 80 + DT_RANK + D_STATE + threadIdx.x];
    __syncthreads();
    const float dtv = dt[(size_t)row * DI + d];
    const float u   = xc[(size_t)row * DI + d];
    const float du  = dtv * u;
    float yv = 0.f;
    #pragma unroll
    for (int n = 0; n < D_STATE; ++n) {
      hc[n] = __expf(dtv * Av[n]) * hc[n] + du * sB[n];
      yv   += hc[n] * sC[n];
    }
    const float zv = xz[(size_t)row * (2 * DI) + DI + d];
    y[(size_t)row * DI + d] = (yv + u * Dd) * siluf(zv);
  }
}

// ---------------------------------------------------------------------------
// Mean over L: out[b,d] = (1/SEQ) * sum_l in[b,l,d]
// ---------------------------------------------------------------------------
__global__ void mean_k(const float* __restrict__ in, float* __restrict__ out)
{
  const int b = blockIdx.x, d = threadIdx.x;
  float s = 0.f;
  for (int l = 0; l < SEQ; ++l)
    s += in[(size_t)(b * SEQ + l) * D_MODEL + d];
  out[b * D_MODEL + d] = s * (1.f / SEQ);
}

// ---------------------------------------------------------------------------
static inline void launch_gemm(const float* A, const float* W, const float* bias,
                               const float* res, float* C, int M, int N, int K,
                               int lda, int ldb, int ldc, int act, hipStream_t stream)
{
  const int tilesM = (M + MREP * 16 - 1) / (MREP * 16);
  const int ntiles = (N + 15) / 16;
  const int waves  = tilesM * ntiles;
  const int blocks = (waves + 7) / 8;            // 8 waves per 256-thread block
  gemm_f32_wmma<<<blocks, 256, 0, stream>>>(A, W, bias, res, C, M, N, K,
                                            lda, ldb, ldc, act);
}

extern "C" void kernel_launch(void* const* d_in, const int* in_sizes, int n_in,
                              void* d_out, int out_size, void* d_ws, size_t ws_size,
                              hipStream_t stream)
{
  const float* x         = (const float*)d_in[0];
  const float* patch_w   = (const float*)d_in[1];
  const float* patch_b   = (const float*)d_in[2];
  const float* in_g      = (const float*)d_in[3];
  const float* in_b      = (const float*)d_in[4];
  const float* ln_g      = (const float*)d_in[5];
  const float* ln_b      = (const float*)d_in[6];
  const float* in_proj_w = (const float*)d_in[7];
  const float* conv_w    = (const float*)d_in[8];
  const float* conv_b    = (const float*)d_in[9];
  const float* x_proj_w  = (const float*)d_in[10];
  const float* dt_proj_w = (const float*)d_in[11];
  const float* dt_proj_b = (const float*)d_in[12];
  const float* A_log     = (const float*)d_in[13];
  const float* Dv        = (const float*)d_in[14];
  const float* m_out_w   = (const float*)d_in[15];
  const float* blk_out_w = (const float*)d_in[16];
  const float* blk_out_b = (const float*)d_in[17];
  const float* fin_g     = (const float*)d_in[18];
  const float* fin_b     = (const float*)d_in[19];
  const float* cls_w     = (const float*)d_in[20];
  const float* cls_b     = (const float*)d_in[21];

  float* ws   = (float*)d_ws;
  float* h    = ws;  ws += (size_t)BL * D_MODEL;      // residual stream
  float* hn   = ws;  ws += (size_t)BL * D_MODEL;      // normed / temp
  float* xz   = ws;  ws += (size_t)BL * 2 * DI;       // in_proj out
  float* xc   = ws;  ws += (size_t)BL * DI;           // conv+silu out
  float* xdbl = ws;  ws += (size_t)BL * 80;           // x_proj out
  float* dtb  = ws;  ws += (size_t)BL * DI;           // softplus(dt)
  float* yb   = ws;  ws += (size_t)BL * DI;           // scan out (gated)
  float* cat  = ws;  ws += (size_t)BL * 2 * D_MODEL;  // [fwd | bwd]
  float* mb   = ws;  ws += (size_t)BSZ * D_MODEL;     // mean pooled

  // Patch embed + input LN
  patch_embed<<<BL, D_MODEL, 0, stream>>>(x, patch_w, patch_b, hn);
  layernorm_k<<<BL, D_MODEL, 0, stream>>>(hn, in_g, in_b, h);

  for (int i = 0; i < N_LAYERS; ++i) {
    layernorm_k<<<BL, D_MODEL, 0, stream>>>(h, ln_g + (size_t)i * D_MODEL,
                                            ln_b + (size_t)i * D_MODEL, hn);
    for (int dir = 0; dir < 2; ++dir) {
      const size_t li = (size_t)i * 2 + dir;
      // xz = hn @ w_in  (2048 x 1024, K=256)
      launch_gemm(hn, in_proj_w + li * D_MODEL * 2 * DI, nullptr, nullptr,
                  xz, BL, 2 * DI, D_MODEL, D_MODEL, 2 * DI, 2 * DI, 0, stream);
      // depthwise conv + SiLU (reversed taps for bwd)
      conv_silu<<<(BL * DI) / 256, 256, 0, stream>>>(
          xz, conv_w + li * DI * D_CONV, conv_b + li * DI, xc, dir);
      // xdbl = xc @ xw   (2048 x 80, K=512)
      launch_gemm(xc, x_proj_w + li * DI * (DT_RANK + 2 * D_STATE), nullptr, nullptr,
                  xdbl, BL, DT_RANK + 2 * D_STATE, DI,
                  DI, DT_RANK + 2 * D_STATE, DT_RANK + 2 * D_STATE, 0, stream);
      // dt = softplus(xdbl[:, :16] @ dtw + dtb)   (2048 x 512, K=16)
      launch_gemm(xdbl, dt_proj_w + li * DT_RANK * DI, dt_proj_b + li * DI, nullptr,
                  dtb, BL, DI, DT_RANK, DT_RANK + 2 * D_STATE, DI, DI, 1, stream);
      // selective scan (+ *silu(z) + u*Dv), direction-aware
      scan_k<<<dim3(DI / 32, BSZ), 32, 0, stream>>>(
          dtb, xc, xdbl, xz, A_log + li * DI * D_STATE, Dv + li * DI, yb, dir);
      // branch output -> cat[:, dir*256 : dir*256+256]   (2048 x 256, K=512)
      launch_gemm(yb, m_out_w + li * DI * D_MODEL, nullptr, nullptr,
                  cat + dir * D_MODEL, BL, D_MODEL, DI, DI, D_MODEL, 2 * D_MODEL,
                  0, stream);
    }
    // h = cat @ blk_out_w + blk_out_b + h   (in-place safe: per-element epilogue)
    launch_gemm(cat, blk_out_w + (size_t)i * 2 * D_MODEL * D_MODEL,
                blk_out_b + (size_t)i * D_MODEL, h,
                h, BL, D_MODEL, 2 * D_MODEL, 2 * D_MODEL, D_MODEL, D_MODEL, 0, stream);
  }

  // Final LN, mean pool, classifier
  layernorm_k<<<BL, D_MODEL, 0, stream>>>(h, fin_g, fin_b, hn);
  mean_k<<<BSZ, D_MODEL, 0, stream>>>(hn, mb);
  launch_gemm(mb, cls_w, cls_b, nullptr, (float*)d_out,
              BSZ, NCLS, D_MODEL, D_MODEL, NCLS, NCLS, 0, stream);
}